// MLLoss_12360915878493
// MI455X (gfx1250) — compile-verified
//
#include <hip/hip_runtime.h>
#include <hip/hip_bf16.h>

// ---------------------------------------------------------------------------
// MLLoss: mean over B of a doctor-label-selected hinge/distance expression.
//   hinge[j] = max(0, 500 - d[j]);  hs = sum_j hinge[j]
//   dl==0 -> d0 + hs - h0 ; dl==1 -> d1 + hs - h1
//   dl==2 -> (rl==0 ? h1 : rl==1 ? h0 : 0) ; else 0
// Memory-bound streaming reduction (~144 MiB -> ~6 us at 23.3 TB/s HBM;
// faster on graph replay since the working set fits the 192 MB L2).
// Per-row math is fully branchless (cmp/cndmask/fma) to avoid exec-mask
// churn under random labels. Wave-level reduction uses
// V_WMMA_F32_16X16X4_F32 (A = lane partials, B = ones): exact in fp32,
// one matrix op per wave, and no ds_bpermute needed.
// ---------------------------------------------------------------------------

typedef float v2f __attribute__((ext_vector_type(2)));
typedef float v8f __attribute__((ext_vector_type(8)));

#define LAMBDA_MARGIN 500.0f
#define THREADS 256
#define WAVES (THREADS / 32)
#define ROWS_PER_THREAD 4

__global__ void mlloss_zero_ws(double* __restrict__ ws) {
    if (threadIdx.x == 0 && blockIdx.x == 0) ws[0] = 0.0;
}

__device__ __forceinline__ float hingef(float d) {
    return fmaxf(0.0f, LAMBDA_MARGIN - d);
}

__global__ void __launch_bounds__(THREADS)
mlloss_reduce(const float4* __restrict__ dist4,   // B x 16 floats = B x 4 float4
              const int*    __restrict__ dlab,
              const int*    __restrict__ rlab,
              double*       __restrict__ ws,
              int Bn) {
    const int tid    = blockIdx.x * blockDim.x + threadIdx.x;
    const int stride = gridDim.x * blockDim.x;

    float acc = 0.0f;
    for (int i = tid; i < Bn; i += stride) {
        const float4 v0 = dist4[i * 4 + 0];
        const float4 v1 = dist4[i * 4 + 1];
        const float4 v2 = dist4[i * 4 + 2];
        const float4 v3 = dist4[i * 4 + 3];

        const float h0 = hingef(v0.x);
        const float h1 = hingef(v0.y);
        float hs = h0 + h1;
        hs += hingef(v0.z); hs += hingef(v0.w);
        hs += hingef(v1.x); hs += hingef(v1.y);
        hs += hingef(v1.z); hs += hingef(v1.w);
        hs += hingef(v2.x); hs += hingef(v2.y);
        hs += hingef(v2.z); hs += hingef(v2.w);
        hs += hingef(v3.x); hs += hingef(v3.y);
        hs += hingef(v3.z); hs += hingef(v3.w);

        const int dli = dlab[i];
        const int rli = rlab[i];

        const float c0 = v0.x + hs - h0;
        const float c1 = v0.y + hs - h1;

        // Branchless selection: masks lower to v_cmp + v_cndmask, the
        // combine lowers to v_fma — no exec-mask manipulation in the loop.
        const float m0 = (dli == 0) ? 1.0f : 0.0f;
        const float m1 = (dli == 1) ? 1.0f : 0.0f;
        const float m2 = (dli == 2) ? 1.0f : 0.0f;
        const float r0 = (rli == 0) ? 1.0f : 0.0f;
        const float r1 = (rli == 1) ? 1.0f : 0.0f;

        const float c2 = r0 * h1 + r1 * h0;
        acc += m0 * c0 + m1 * c1 + m2 * c2;
    }

    // ---- wave32 reduction via V_WMMA_F32_16X16X4_F32 ----------------------
    // A (16x4): lane L<16 holds {A[L][0], A[L][1]}, lane L>=16 holds
    // {A[L-16][2], A[L-16][3]}. Put acc in component 0, zero component 1.
    // B = all ones.  D[m][n] = acc[m] + acc[m+16]  (same for every n).
    v2f av; av[0] = acc;  av[1] = 0.0f;
    v2f bv; bv[0] = 1.0f; bv[1] = 1.0f;
    v8f cv = {};
    cv = __builtin_amdgcn_wmma_f32_16x16x4_f32(
            /*neg_a=*/false, av, /*neg_b=*/false, bv,
            /*c_mod=*/(short)0, cv, /*reuse_a=*/false, /*reuse_b=*/false);

    // Lanes 0-15 uniformly hold sum(acc[0..7],acc[16..23]);
    // lanes 16-31 uniformly hold sum(acc[8..15],acc[24..31]).
    // Lane 0 and lane 16 each deposit their half-sum — no shuffle needed.
    const float s = cv[0] + cv[1] + cv[2] + cv[3] + cv[4] + cv[5] + cv[6] + cv[7];

    __shared__ float wave_sum[WAVES * 2];
    const int lane = threadIdx.x & 31;
    const int wave = threadIdx.x >> 5;
    if ((lane & 15) == 0) wave_sum[wave * 2 + (lane >> 4)] = s;
    __syncthreads();

    if (threadIdx.x == 0) {
        float t = 0.0f;
        #pragma unroll
        for (int w = 0; w < WAVES * 2; ++w) t += wave_sum[w];
        atomicAdd(ws, (double)t);   // GLOBAL_ATOMIC_ADD_F64
    }
}

__global__ void mlloss_finalize(const double* __restrict__ ws,
                                float* __restrict__ out, int Bn) {
    if (threadIdx.x == 0 && blockIdx.x == 0)
        out[0] = (float)(ws[0] / (double)Bn);
}

extern "C" void kernel_launch(void* const* d_in, const int* in_sizes, int n_in,
                              void* d_out, int out_size, void* d_ws, size_t ws_size,
                              hipStream_t stream) {
    const float4* dist4 = (const float4*)d_in[0];   // (B,16) fp32
    const int*    dlab  = (const int*)d_in[1];      // (B,)  int
    const int*    rlab  = (const int*)d_in[2];      // (B,)  int
    float*        out   = (float*)d_out;
    double*       ws    = (double*)d_ws;

    const int Bn = in_sizes[1];                     // rows
    int blocks = (Bn + THREADS * ROWS_PER_THREAD - 1) / (THREADS * ROWS_PER_THREAD);
    if (blocks < 1) blocks = 1;

    mlloss_zero_ws<<<1, 1, 0, stream>>>(ws);
    mlloss_reduce<<<blocks, THREADS, 0, stream>>>(dist4, dlab, rlab, ws, Bn);
    mlloss_finalize<<<1, 1, 0, stream>>>(ws, out, Bn);
}